// L2ChebConv_84859963834418
// MI455X (gfx1250) — compile-verified
//
#include <hip/hip_runtime.h>
#include <hip/hip_bf16.h>

// ---------------------------------------------------------------------------
// ChebConv x2 on MI455X (gfx1250, wave32, WMMA f32 16x16x4)
//   layer1: X4 = [T0|T1|T2|T3] (N x 256) fp32, GEMM vs stacked W1 (256 x 400)
//   layer2: scatter-recurrence on [N,400] + float4 dot to 4 outputs
// ---------------------------------------------------------------------------

typedef __attribute__((ext_vector_type(2))) float v2f;
typedef __attribute__((ext_vector_type(8))) float v8f;

#define IN_C  64
#define HID   400
#define OUT_C 4
#define XK    256   // 4 * IN_C
#define LDA   260   // padded LDS stride (bank-conflict free: 260 % 64 == 4)

// ---------------- small utility kernels ----------------

__global__ void k_zero(float* __restrict__ p, int n) {
    int i = blockIdx.x * blockDim.x + threadIdx.x;
    if (i < n) p[i] = 0.0f;
}

__global__ void k_deg(const int* __restrict__ src, float* __restrict__ deg, int E) {
    int e = blockIdx.x * blockDim.x + threadIdx.x;
    if (e < E) atomicAdd(&deg[src[e]], 1.0f);
}

__global__ void k_dinv(const float* __restrict__ deg, float* __restrict__ dinv, int n) {
    int i = blockIdx.x * blockDim.x + threadIdx.x;
    if (i < n) {
        float d = deg[i];
        dinv[i] = (d > 0.0f) ? rsqrtf(d) : 0.0f;
    }
}

__global__ void k_norm(const int* __restrict__ src, const int* __restrict__ dst,
                       const float* __restrict__ dinv, float* __restrict__ nrm, int E) {
    int e = blockIdx.x * blockDim.x + threadIdx.x;
    if (e < E) nrm[e] = -dinv[src[e]] * dinv[dst[e]];
}

// out[node*ldo+offo+c] = in ? scale*in[node*ldi+offi+c] : 0      (c in [0,C))
__global__ void k_init_block(float* __restrict__ out, int ldo, int offo,
                             const float* __restrict__ in, int ldi, int offi,
                             int C, long long total, float scale) {
    long long t = (long long)blockIdx.x * blockDim.x + threadIdx.x;
    if (t >= total) return;
    int node = (int)(t / C);
    int c    = (int)(t % C);
    float v = 0.0f;
    if (in) v = scale * in[(size_t)node * ldi + offi + c];
    out[(size_t)node * ldo + offo + c] = v;
}

// hout[dst[e]] += coef*norm[e]*hin[src[e]]   (4 channels per thread, float4 gather)
__global__ void k_scatter(const int* __restrict__ src, const int* __restrict__ dst,
                          const float* __restrict__ nrm,
                          const float* __restrict__ hin, int ldi, int offi,
                          float* __restrict__ hout, int ldo, int offo,
                          int Cq, float coef, long long total) {
    long long t = (long long)blockIdx.x * blockDim.x + threadIdx.x;
    if (t >= total) return;
    int e = (int)(t / Cq);
    int c = (int)(t % Cq) * 4;
    float w = coef * nrm[e];
    int s = src[e], d = dst[e];
    const float4 v = *(const float4*)(hin + (size_t)s * ldi + offi + c);
    float* o = hout + (size_t)d * ldo + offo + c;
    atomicAdd(o + 0, w * v.x);
    atomicAdd(o + 1, w * v.y);
    atomicAdd(o + 2, w * v.z);
    atomicAdd(o + 3, w * v.w);
}

// ---------------- layer-1 GEMM: H = relu(X4 @ W + b) with WMMA f32 16x16x4 ----------------
// Block = 16 output rows x all 400 cols. 8 waves/block; wave w owns N-tiles
// {w, w+8, w+16} plus a 4th tile clamped to tile 24 (stored only by wave 0, so
// the K loop is branch-free and EXEC is trivially all-ones for every WMMA).

__device__ __forceinline__ void store_tile(const v8f& acc, int m0, int n0, int l15, int kh,
                                           const float* __restrict__ bias,
                                           float* __restrict__ H) {
    const float bv = bias[n0 + l15];
    #pragma unroll
    for (int r = 0; r < 8; ++r) {
        int m = m0 + r + 8 * kh;           // C/D layout: VGPR r -> M=r (lanes 0-15), M=r+8 (16-31)
        float v = acc[r] + bv;
        H[(size_t)m * HID + n0 + l15] = v > 0.0f ? v : 0.0f;
    }
}

__launch_bounds__(256)
__global__ void k_gemm1(const float* __restrict__ X4, const float* __restrict__ W,
                        const float* __restrict__ bias, float* __restrict__ H) {
    __shared__ float As[16 * LDA];
    const int m0 = blockIdx.x * 16;

    // cooperative load of the contiguous 16x256 fp32 A tile (float4 granularity)
    for (int idx = threadIdx.x; idx < 16 * (XK / 4); idx += 256) {
        int r  = idx / (XK / 4);
        int c4 = (idx % (XK / 4)) * 4;
        const float4 v = *(const float4*)(X4 + (size_t)(m0 + r) * XK + c4);
        float* p = &As[r * LDA + c4];
        p[0] = v.x; p[1] = v.y; p[2] = v.z; p[3] = v.w;
    }
    __syncthreads();

    const int wave = threadIdx.x >> 5;
    const int lane = threadIdx.x & 31;
    const int l15  = lane & 15;   // M for A, N for B/C/D
    const int kh   = lane >> 4;   // half-select
    const int n0a = (wave)      * 16;
    const int n0b = (wave + 8)  * 16;
    const int n0c = (wave + 16) * 16;
    // 25 N-tiles total: only wave 0 owns a 4th tile (24). All other waves
    // redundantly compute tile 24 too (branch-free hot loop), but don't store.
    const int n0d = 24 * 16;

    v8f acc0 = {}, acc1 = {}, acc2 = {}, acc3 = {};

    #pragma unroll 4
    for (int kk = 0; kk < XK; kk += 4) {
        // A fragment (16x4 f32): lane<16 holds K=kk,kk+1 ; lane>=16 holds K=kk+2,kk+3
        v2f a;
        a.x = As[l15 * LDA + kk + 2 * kh];
        a.y = As[l15 * LDA + kk + 2 * kh + 1];
        const int krow = kk + 2 * kh;

        v2f b;
        b.x = W[(size_t)krow * HID + n0a + l15];
        b.y = W[(size_t)(krow + 1) * HID + n0a + l15];
        acc0 = __builtin_amdgcn_wmma_f32_16x16x4_f32(false, a, false, b, (short)0, acc0, false, false);

        b.x = W[(size_t)krow * HID + n0b + l15];
        b.y = W[(size_t)(krow + 1) * HID + n0b + l15];
        acc1 = __builtin_amdgcn_wmma_f32_16x16x4_f32(false, a, false, b, (short)0, acc1, false, false);

        b.x = W[(size_t)krow * HID + n0c + l15];
        b.y = W[(size_t)(krow + 1) * HID + n0c + l15];
        acc2 = __builtin_amdgcn_wmma_f32_16x16x4_f32(false, a, false, b, (short)0, acc2, false, false);

        b.x = W[(size_t)krow * HID + n0d + l15];
        b.y = W[(size_t)(krow + 1) * HID + n0d + l15];
        acc3 = __builtin_amdgcn_wmma_f32_16x16x4_f32(false, a, false, b, (short)0, acc3, false, false);
    }

    store_tile(acc0, m0, n0a, l15, kh, bias, H);
    store_tile(acc1, m0, n0b, l15, kh, bias, H);
    store_tile(acc2, m0, n0c, l15, kh, bias, H);
    if (wave == 0) store_tile(acc3, m0, n0d, l15, kh, bias, H);  // epilogue-only branch
}

// ---------------- layer-2 projection: out(+)= T @ W2k (+b2) (+relu) ----------------
// mode 0: out = b2 + T@W2k ; mode 1: out += T@W2k ; mode 2: out = relu(out + T@W2k)

__global__ void k_dot2(const float* __restrict__ T, const float* __restrict__ W2k,
                       const float* __restrict__ b2, float* __restrict__ out,
                       int n, int mode) {
    int i = blockIdx.x * blockDim.x + threadIdx.x;
    if (i >= n) return;
    float4 acc = (mode == 0) ? *(const float4*)b2
                             : *(const float4*)(out + (size_t)i * 4);
    const float* trow = T + (size_t)i * HID;
    for (int c = 0; c < HID; c += 4) {
        const float4 tv = *(const float4*)(trow + c);
        const float4 w0 = *(const float4*)(W2k + (size_t)(c + 0) * 4);
        const float4 w1 = *(const float4*)(W2k + (size_t)(c + 1) * 4);
        const float4 w2 = *(const float4*)(W2k + (size_t)(c + 2) * 4);
        const float4 w3 = *(const float4*)(W2k + (size_t)(c + 3) * 4);
        acc.x += tv.x * w0.x + tv.y * w1.x + tv.z * w2.x + tv.w * w3.x;
        acc.y += tv.x * w0.y + tv.y * w1.y + tv.z * w2.y + tv.w * w3.y;
        acc.z += tv.x * w0.z + tv.y * w1.z + tv.z * w2.z + tv.w * w3.z;
        acc.w += tv.x * w0.w + tv.y * w1.w + tv.z * w2.w + tv.w * w3.w;
    }
    if (mode == 2) {
        acc.x = acc.x > 0.f ? acc.x : 0.f;
        acc.y = acc.y > 0.f ? acc.y : 0.f;
        acc.z = acc.z > 0.f ? acc.z : 0.f;
        acc.w = acc.w > 0.f ? acc.w : 0.f;
    }
    *(float4*)(out + (size_t)i * 4) = acc;
}

// ---------------- host-side orchestration ----------------

extern "C" void kernel_launch(void* const* d_in, const int* in_sizes, int n_in,
                              void* d_out, int out_size, void* d_ws, size_t ws_size,
                              hipStream_t stream) {
    const float* x  = (const float*)d_in[0];
    const int*   ei = (const int*)d_in[1];
    const float* W1 = (const float*)d_in[2];
    const float* b1 = (const float*)d_in[3];
    const float* W2 = (const float*)d_in[4];
    const float* b2 = (const float*)d_in[5];
    float* out = (float*)d_out;

    const int N = in_sizes[0] / IN_C;   // 50000
    const int E = in_sizes[1] / 2;      // 800000
    const int* src = ei;
    const int* dst = ei + E;

    // workspace carve-out (256B aligned)
    char* base = (char*)d_ws;
    size_t off = 0;
    auto carve = [&](size_t bytes) {
        void* p = base + off;
        off = (off + bytes + 255) & ~(size_t)255;
        return p;
    };
    float* deg  = (float*)carve((size_t)N * 4);
    float* dinv = (float*)carve((size_t)N * 4);
    float* nrm  = (float*)carve((size_t)E * 4);
    float* X4   = (float*)carve((size_t)N * XK * 4);    // 51.2 MB
    float* H    = (float*)carve((size_t)N * HID * 4);   // 80 MB
    float* Bu1  = (float*)carve((size_t)N * HID * 4);   // 80 MB
    float* Bu2  = (float*)carve((size_t)N * HID * 4);   // 80 MB
    (void)ws_size; (void)n_in; (void)out_size;

    const int TB = 256;
    auto nb = [&](long long t) { return dim3((unsigned)((t + TB - 1) / TB)); };

    // normalization
    k_zero<<<nb(N), TB, 0, stream>>>(deg, N);
    k_deg <<<nb(E), TB, 0, stream>>>(src, deg, E);
    k_dinv<<<nb(N), TB, 0, stream>>>(deg, dinv, N);
    k_norm<<<nb(E), TB, 0, stream>>>(src, dst, dinv, nrm, E);

    // ---- layer 1: build X4 = [T0|T1|T2|T3] (column blocks of 64) ----
    const long long nc1 = (long long)N * IN_C;
    const long long sc1 = (long long)E * (IN_C / 4);
    k_init_block<<<nb(nc1), TB, 0, stream>>>(X4, XK, 0,   x,  IN_C, 0, IN_C, nc1, 1.0f);   // T0 = x
    k_init_block<<<nb(nc1), TB, 0, stream>>>(X4, XK, 64,  nullptr, 0, 0, IN_C, nc1, 0.0f); // T1 = 0
    k_scatter   <<<nb(sc1), TB, 0, stream>>>(src, dst, nrm, X4, XK, 0,   X4, XK, 64,  IN_C/4, 1.0f, sc1);
    k_init_block<<<nb(nc1), TB, 0, stream>>>(X4, XK, 128, X4, XK, 0,  IN_C, nc1, -1.0f);   // T2 = -T0 ...
    k_scatter   <<<nb(sc1), TB, 0, stream>>>(src, dst, nrm, X4, XK, 64,  X4, XK, 128, IN_C/4, 2.0f, sc1); // ... + 2 prop(T1)
    k_init_block<<<nb(nc1), TB, 0, stream>>>(X4, XK, 192, X4, XK, 64, IN_C, nc1, -1.0f);   // T3 = -T1 ...
    k_scatter   <<<nb(sc1), TB, 0, stream>>>(src, dst, nrm, X4, XK, 128, X4, XK, 192, IN_C/4, 2.0f, sc1); // ... + 2 prop(T2)

    // H = relu(X4 @ W1cat + b1)   (N % 16 == 0, HID = 25*16, XK = 64*4)
    k_gemm1<<<dim3((unsigned)(N / 16)), TB, 0, stream>>>(X4, W1, b1, H);

    // ---- layer 2: recurrence on [N, 400] with fused output accumulation ----
    const long long nc2 = (long long)N * HID;
    const long long sc2 = (long long)E * (HID / 4);
    k_dot2      <<<nb(N),   TB, 0, stream>>>(H,   W2 + 0 * (HID * OUT_C), b2, out, N, 0); // out = b2 + T0@W2[0]
    k_init_block<<<nb(nc2), TB, 0, stream>>>(Bu1, HID, 0, nullptr, 0, 0, HID, nc2, 0.0f);
    k_scatter   <<<nb(sc2), TB, 0, stream>>>(src, dst, nrm, H,   HID, 0, Bu1, HID, 0, HID/4, 1.0f, sc2); // T1
    k_dot2      <<<nb(N),   TB, 0, stream>>>(Bu1, W2 + 1 * (HID * OUT_C), b2, out, N, 1);
    k_init_block<<<nb(nc2), TB, 0, stream>>>(Bu2, HID, 0, H, HID, 0, HID, nc2, -1.0f);
    k_scatter   <<<nb(sc2), TB, 0, stream>>>(src, dst, nrm, Bu1, HID, 0, Bu2, HID, 0, HID/4, 2.0f, sc2); // T2
    k_dot2      <<<nb(N),   TB, 0, stream>>>(Bu2, W2 + 2 * (HID * OUT_C), b2, out, N, 1);
    k_init_block<<<nb(nc2), TB, 0, stream>>>(Bu1, HID, 0, Bu1, HID, 0, HID, nc2, -1.0f);   // in-place negate T1
    k_scatter   <<<nb(sc2), TB, 0, stream>>>(src, dst, nrm, Bu2, HID, 0, Bu1, HID, 0, HID/4, 2.0f, sc2); // T3
    k_dot2      <<<nb(N),   TB, 0, stream>>>(Bu1, W2 + 3 * (HID * OUT_C), b2, out, N, 2); // relu finish
}